// DistributedGCNConv_4440996184259
// MI455X (gfx1250) — compile-verified
//
#include <hip/hip_runtime.h>

typedef float v2f __attribute__((ext_vector_type(2)));
typedef float v8f __attribute__((ext_vector_type(8)));

#define C_DIM 128   // IN_C == OUT_C == 128

// ---------------------------------------------------------------------------
// Zero the accumulator (d_out) before edge scatter.
// ---------------------------------------------------------------------------
__global__ void gcn_zero_kernel(float* __restrict__ p, int n) {
    int i = blockIdx.x * blockDim.x + threadIdx.x;
    if (i < n) p[i] = 0.0f;
}

// ---------------------------------------------------------------------------
// h[m,:] = dinv[m] * (x[m,:] @ W)   via V_WMMA_F32_16X16X4_F32
// Block = 256 threads = 8 waves. W^T staged in LDS (64 KB), reused by all
// waves. Each wave computes one 16-row M tile across all 8 N tiles, so each
// A fragment (2 fp32/lane) feeds 8 WMMAs. 32 K-steps * 8 N tiles = 256 WMMA.
// ---------------------------------------------------------------------------
__global__ __launch_bounds__(256) void gcn_gemm_wmma_kernel(
    const float* __restrict__ x, const float* __restrict__ w,
    const float* __restrict__ dinv, float* __restrict__ h, int n_rows) {
    // wt[n*128 + k] = w[k*128 + n]  (W transposed -> contiguous K per lane)
    __shared__ float wt[C_DIM * C_DIM];
    {
        int t = threadIdx.x;
        #pragma unroll
        for (int i = 0; i < (C_DIM * C_DIM) / 256; ++i) {
            int idx = t + i * 256;            // coalesced global read
            int k = idx >> 7, n = idx & 127;
            wt[n * C_DIM + k] = w[idx];
        }
    }
    __syncthreads();

    const int wave   = threadIdx.x >> 5;
    const int lane   = threadIdx.x & 31;
    const int m_tile = blockIdx.x * 8 + wave;
    const int n_mt   = n_rows >> 4;           // 100000/16 = 6250 tiles
    if (m_tile >= n_mt) return;               // whole-wave exit: EXEC all-1 for WMMA

    const int half = lane >> 4;               // K-pair / M-half selector
    const int nl   = lane & 15;               // N (or M) index within tile
    const int arow = m_tile * 16 + nl;        // A-matrix row for this lane

    v8f acc[8] = {};                          // 8 N tiles of 16x16 fp32 C/D

    const float* __restrict__ xrow = x + (size_t)arow * C_DIM;
    #pragma unroll 4
    for (int k = 0; k < C_DIM; k += 4) {
        // A 16x4 f32 layout: lanes 0-15 -> K={k,k+1}; lanes 16-31 -> K={k+2,k+3}
        v2f av = *(const v2f*)(xrow + k + 2 * half);
        #pragma unroll
        for (int tt = 0; tt < 8; ++tt) {
            // B 4x16 f32 layout (from W^T in LDS): contiguous K pair per lane
            v2f bv = *(const v2f*)(wt + (tt * 16 + nl) * C_DIM + k + 2 * half);
            acc[tt] = __builtin_amdgcn_wmma_f32_16x16x4_f32(
                false, av, false, bv, (short)0, acc[tt], false, false);
        }
    }

    // C/D layout: VGPR r -> M = r + 8*half, N = nl (per N tile). Scale by dinv.
    #pragma unroll
    for (int r = 0; r < 8; ++r) {
        int mrow = m_tile * 16 + r + 8 * half;
        float d  = dinv[mrow];
        float* __restrict__ hp = h + (size_t)mrow * C_DIM + nl;
        #pragma unroll
        for (int tt = 0; tt < 8; ++tt)
            hp[tt * 16] = acc[tt][r] * d;
    }
}

// ---------------------------------------------------------------------------
// Edge scatter: one wave per edge. Lane l gathers h[col[e], 4l..4l+3] as one
// b128 load (512 B coalesced per wave, L2-resident) and does 4 fp32 atomics
// into out[row[e], :]. Edges with row >= n_owned are dropped (reference keeps
// only agg[:num_owned]).
// ---------------------------------------------------------------------------
__global__ __launch_bounds__(256) void gcn_scatter_kernel(
    const float* __restrict__ h, const int* __restrict__ row,
    const int* __restrict__ col, float* __restrict__ out,
    int n_edges, int n_owned) {
    int e = blockIdx.x * 8 + (threadIdx.x >> 5);
    if (e >= n_edges) return;
    int r = row[e];                           // uniform per wave -> broadcast
    if (r >= n_owned) return;
    int c = col[e];
    int lane = threadIdx.x & 31;
    const float4 v = *(const float4*)(h + (size_t)c * C_DIM + lane * 4);
    float* o = out + (size_t)r * C_DIM + lane * 4;
    unsafeAtomicAdd(o + 0, v.x);              // global_atomic_add_f32
    unsafeAtomicAdd(o + 1, v.y);
    unsafeAtomicAdd(o + 2, v.z);
    unsafeAtomicAdd(o + 3, v.w);
}

// ---------------------------------------------------------------------------
// out[i,c] = out[i,c] * dinv[i] + bias[c]   (in-place finalize over owned rows)
// ---------------------------------------------------------------------------
__global__ void gcn_finalize_kernel(float* __restrict__ out,
                                    const float* __restrict__ dinv,
                                    const float* __restrict__ bias, int n) {
    int i = blockIdx.x * blockDim.x + threadIdx.x;
    if (i < n) {
        int r = i >> 7, c = i & 127;
        out[i] = out[i] * dinv[r] + bias[c];
    }
}

extern "C" void kernel_launch(void* const* d_in, const int* in_sizes, int n_in,
                              void* d_out, int out_size, void* d_ws, size_t ws_size,
                              hipStream_t stream) {
    const float* x    = (const float*)d_in[0];   // [N_local, 128]
    const float* w    = (const float*)d_in[1];   // [128, 128]
    const float* bias = (const float*)d_in[2];   // [128]
    const float* dinv = (const float*)d_in[3];   // [N_local]
    const int*   row  = (const int*)d_in[4];     // [E] dst
    const int*   col  = (const int*)d_in[5];     // [E] src
    // d_in[6] = num_owned device scalar; derive from out_size (capture-safe)

    const int n_local = in_sizes[0] / C_DIM;     // 100000
    const int n_edges = in_sizes[4];             // 1600000
    const int n_owned = out_size / C_DIM;        // 80000

    float* out = (float*)d_out;
    float* h   = (float*)d_ws;                   // n_local*128 fp32 = 51.2 MB

    // 1) zero accumulator
    gcn_zero_kernel<<<(out_size + 255) / 256, 256, 0, stream>>>(out, out_size);

    // 2) WMMA GEMM + source-side degree scale
    int m_tiles = n_local / 16;
    gcn_gemm_wmma_kernel<<<(m_tiles + 7) / 8, 256, 0, stream>>>(x, w, dinv, h, n_local);

    // 3) edge gather + atomic scatter (owned destinations only)
    gcn_scatter_kernel<<<(n_edges + 7) / 8, 256, 0, stream>>>(h, row, col, out,
                                                              n_edges, n_owned);

    // 4) destination-side degree scale + bias, in place
    gcn_finalize_kernel<<<(out_size + 255) / 256, 256, 0, stream>>>(out, dinv, bias,
                                                                    out_size);
}